// Quantize_26740466384921
// MI455X (gfx1250) — compile-verified
//
#include <hip/hip_runtime.h>
#include <hip/hip_bf16.h>

typedef __attribute__((ext_vector_type(2))) float v2f;
typedef __attribute__((ext_vector_type(8))) float v8f;

#define BATCH 16384
#define CHANS 1024
#define GROUPS 4
#define DDIM 256            // CHANS / GROUPS
#define NEMB 1024
#define NROWS (BATCH * GROUPS)   // 65536
#define ROWS_PER_BLOCK 16
#define NWAVES 8
#define TILES_PER_WAVE 8         // 64 N-tiles of 16 codes / 8 waves
#define NBLOCKS (NROWS / ROWS_PER_BLOCK)  // 4096

// ---------------------------------------------------------------- norms
__global__ __launch_bounds__(256) void vq_norms(const float* __restrict__ embed,
                                                float* __restrict__ enorm) {
    int j = blockIdx.x * blockDim.x + threadIdx.x;
    if (j < NEMB) {
        const float4* e = (const float4*)(embed + (long)j * DDIM);
        float s = 0.f;
        #pragma unroll 4
        for (int d = 0; d < DDIM / 4; ++d) {
            float4 v = e[d];
            s += v.x * v.x + v.y * v.y + v.z * v.z + v.w * v.w;
        }
        enorm[j] = s;
    }
}

// ---------------------------------------------------------------- main VQ
__global__ __launch_bounds__(256) void vq_main(const float* __restrict__ z,
                                               const float* __restrict__ embed,
                                               const float* __restrict__ enorm,
                                               float* __restrict__ out,
                                               float* __restrict__ partial) {
    __shared__ float zt[ROWS_PER_BLOCK * DDIM];     // 16 KB z tile
    __shared__ float redV[NWAVES][ROWS_PER_BLOCK];
    __shared__ int   redI[NWAVES][ROWS_PER_BLOCK];
    __shared__ int   finalIdx[ROWS_PER_BLOCK];
    __shared__ float rsum[256];

    const int tid  = threadIdx.x;
    const int wave = tid >> 5;
    const int lane = tid & 31;
    const long rowBase = (long)blockIdx.x * ROWS_PER_BLOCK;

    // stage z tile to LDS (4096 floats, 256 threads, float4 vectorized)
    {
        const float4* src = (const float4*)(z + rowBase * DDIM);
        float4* dst = (float4*)zt;
        for (int i = tid; i < (ROWS_PER_BLOCK * DDIM) / 4; i += 256) dst[i] = src[i];
    }
    __syncthreads();

    const int halfSel = lane >> 4;      // 0: K={0,1}, 1: K={2,3}
    const int l16     = lane & 15;
    const int kOff    = halfSel * 2;

    float best[8];
    int   bidx[8];
    #pragma unroll
    for (int r = 0; r < 8; ++r) { best[r] = 3.4e38f; bidx[r] = 0x7fffffff; }

    const float* arow = zt + l16 * DDIM;  // A fragment row (M = l16) in LDS

    for (int t = 0; t < TILES_PER_WAVE; ++t) {
        const int nBase = (wave * TILES_PER_WAVE + t) * 16;
        const int n     = nBase + l16;                 // B fragment column
        const float* brow = embed + (long)n * DDIM;    // codebook row (L2-resident)

        v8f acc = {};
        #pragma unroll 4
        for (int k = 0; k < DDIM; k += 4) {
            // ISA 7.12.2 fp32 16x4 A / 4x16 B layout:
            // lanes 0-15 hold K=k,k+1 ; lanes 16-31 hold K=k+2,k+3
            v2f a = *(const v2f*)(arow + k + kOff);
            v2f b = *(const v2f*)(brow + k + kOff);
            acc = __builtin_amdgcn_wmma_f32_16x16x4_f32(
                false, a, false, b, (short)0, acc, false, false);
        }

        const float en = enorm[n];
        #pragma unroll
        for (int r = 0; r < 8; ++r) {
            // dist - ||z||^2 = ||e||^2 - 2 z.e  (row-constant term irrelevant to argmin)
            float s = en - 2.0f * acc[r];
            if (s < best[r]) { best[r] = s; bidx[r] = n; }   // strict < keeps lowest n
        }
    }

    // reduce across the 16 lanes sharing each row (masks < 16 stay in half)
    #pragma unroll
    for (int r = 0; r < 8; ++r) {
        float v = best[r]; int ix = bidx[r];
        #pragma unroll
        for (int m = 1; m < 16; m <<= 1) {
            float ov = __shfl_xor(v, m, 32);
            int   oi = __shfl_xor(ix, m, 32);
            if (ov < v || (ov == v && oi < ix)) { v = ov; ix = oi; }
        }
        best[r] = v; bidx[r] = ix;
    }
    // lane 0 -> rows 0..7 ; lane 16 -> rows 8..15
    if (l16 == 0) {
        #pragma unroll
        for (int r = 0; r < 8; ++r) {
            int row = r + 8 * halfSel;
            redV[wave][row] = best[r];
            redI[wave][row] = bidx[r];
        }
    }
    __syncthreads();

    // cross-wave reduce in fixed wave order (waves own increasing n ranges)
    if (tid < ROWS_PER_BLOCK) {
        float v = redV[0][tid]; int ix = redI[0][tid];
        #pragma unroll
        for (int w = 1; w < NWAVES; ++w) {
            float ov = redV[w][tid]; int oi = redI[w][tid];
            if (ov < v || (ov == v && oi < ix)) { v = ov; ix = oi; }
        }
        finalIdx[tid] = ix;
        out[(long)BATCH * CHANS + 1 + rowBase + tid] = (float)ix;  // ind output
    }
    __syncthreads();

    // write z_q = embed[ind] and accumulate squared error (deterministic)
    float ss = 0.f;
    {
        const int row = tid >> 4;
        const int d0  = (tid & 15) * 16;
        const float* eq = embed + (long)finalIdx[row] * DDIM + d0;
        const float* ze = zt + row * DDIM + d0;
        float* o = out + (rowBase + row) * DDIM + d0;
        #pragma unroll
        for (int d = 0; d < 16; ++d) {
            float q = eq[d];
            float df = q - ze[d];
            ss += df * df;
            o[d] = q;
        }
    }
    rsum[tid] = ss;
    __syncthreads();
    for (int s = 128; s > 0; s >>= 1) {
        if (tid < s) rsum[tid] += rsum[tid + s];
        __syncthreads();
    }
    if (tid == 0) partial[blockIdx.x] = rsum[0];
}

// ---------------------------------------------------------------- finish
__global__ __launch_bounds__(256) void vq_finish(const float* __restrict__ partial,
                                                 float* __restrict__ out) {
    __shared__ float r[256];
    float s = 0.f;
    for (int i = threadIdx.x; i < NBLOCKS; i += 256) s += partial[i];  // fixed order
    r[threadIdx.x] = s;
    __syncthreads();
    for (int st = 128; st > 0; st >>= 1) {
        if (threadIdx.x < st) r[threadIdx.x] += r[threadIdx.x + st];
        __syncthreads();
    }
    if (threadIdx.x == 0) {
        // diff = 10 * (0.25*mse + mse) = 12.5 * mse, mse over NROWS*DDIM elements
        out[(long)BATCH * CHANS] = r[0] * (12.5f / (float)((long)NROWS * DDIM));
    }
}

extern "C" void kernel_launch(void* const* d_in, const int* in_sizes, int n_in,
                              void* d_out, int out_size, void* d_ws, size_t ws_size,
                              hipStream_t stream) {
    const float* z     = (const float*)d_in[0];
    const float* embed = (const float*)d_in[1];
    float* out = (float*)d_out;

    float* enorm   = (float*)d_ws;        // 1024 floats
    float* partial = enorm + NEMB;        // 4096 floats  (20 KB total scratch)

    vq_norms<<<(NEMB + 255) / 256, 256, 0, stream>>>(embed, enorm);
    vq_main<<<NBLOCKS, 256, 0, stream>>>(z, embed, enorm, out, partial);
    vq_finish<<<1, 256, 0, stream>>>(partial, out);
}